// NonParameteric_19164144075394
// MI455X (gfx1250) — compile-verified
//
#include <hip/hip_runtime.h>
#include <hip/hip_bf16.h>

// Problem constants (match reference)
#define BB 256   // batch
#define SS 256   // seq len
#define HH 768   // hidden
#define H4 192   // HH / 4 (float4 columns)
#define CC 150   // class prototypes
#define CPAD 160 // C padded to multiple of 16 for WMMA tiles

typedef float v2f __attribute__((ext_vector_type(2)));
typedef float v8f __attribute__((ext_vector_type(8)));

// ---------------------------------------------------------------------------
// Kernel 1: masked mean pooling  [B,S,H] -> [B,H]
// One block per batch row, 192 threads; each thread owns one float4 column
// (16B), streaming S rows with global_load_b128 — widest VMEM for the
// bandwidth-bound 201 MB read that dominates this op.
// ---------------------------------------------------------------------------
__global__ void pool_kernel(const float4* __restrict__ lhs4,  // [B,S,H/4]
                            const int* __restrict__ mask,     // [B,S]
                            float4* __restrict__ pre4) {      // [B,H/4]
    const int b = blockIdx.x;
    const int t = threadIdx.x;                 // 0..191
    const float4* base = lhs4 + (size_t)b * SS * H4 + t;
    const int* mrow = mask + (size_t)b * SS;

    float ax = 0.f, ay = 0.f, az = 0.f, aw = 0.f, cnt = 0.f;
    for (int s = 0; s < SS; ++s) {
        const float m = (float)mrow[s];
        cnt += m;
        const float4 v = base[(size_t)s * H4];
        ax += m * v.x;
        ay += m * v.y;
        az += m * v.z;
        aw += m * v.w;
    }
    const float inv = 1.0f / fmaxf(cnt, 1e-9f);
    float4 o;
    o.x = ax * inv; o.y = ay * inv; o.z = az * inv; o.w = aw * inv;
    pre4[(size_t)b * H4 + t] = o;
}

// ---------------------------------------------------------------------------
// Kernel 2: t_m[c] = means[c] @ cov_inv @ means[c]^T
// One block per class, 768 threads; thread j computes (means[c] @ cov_inv)[j]
// then a shared-memory tree reduce (768 = 512+256 two-phase).
// cov_inv (2.25 MB) is L2-resident across the 150 blocks.
// ---------------------------------------------------------------------------
__global__ void tm_kernel(const float* __restrict__ means,
                          const float* __restrict__ covinv,
                          float* __restrict__ t_m) {
    __shared__ float red[768];
    const int c = blockIdx.x;
    const int j = threadIdx.x;                 // 0..767
    const float* mu = means + (size_t)c * HH;

    float acc = 0.f;
    for (int i = 0; i < HH; ++i)
        acc += mu[i] * covinv[(size_t)i * HH + j];
    red[j] = acc * mu[j];
    __syncthreads();

    if (j < 256) red[j] += red[j + 512];       // fold 768 -> 512
    __syncthreads();
    for (int off = 256; off >= 1; off >>= 1) {
        if (j < off) red[j] += red[j + off];
        __syncthreads();
    }
    if (j == 0) t_m[c] = red[0];
}

// ---------------------------------------------------------------------------
// Kernel 3: xS = prelogits @ cov_inv  ([256,768] x [768,768], fp32 WMMA)
// One wave32 per 16x16 output tile; K-loop of 192 v_wmma_f32_16x16x4_f32.
// ISA f32 layouts: A 16x4 -> lane<16 holds (M=lane, K={0,1}), lane>=16 holds
// (M=lane-16, K={2,3}); B 4x16 -> VGPR0/1 rows K={0,2}/{1,3}, N=lane%16;
// D VGPR v -> row v (lanes 0-15) or v+8 (lanes 16-31), col = lane%16.
// ---------------------------------------------------------------------------
__global__ void xs_wmma_kernel(const float* __restrict__ A,   // [B,H]
                               const float* __restrict__ Bm,  // [H,H]
                               float* __restrict__ D) {       // [B,H]
    const int tile = blockIdx.x;               // 16 * 48 tiles
    const int tm = tile / (HH / 16);
    const int tn = tile % (HH / 16);
    const int lane = threadIdx.x;              // 0..31
    const int row  = lane & 15;
    const int khi  = lane >> 4;                // 0 or 1
    const int m0 = tm * 16, n0 = tn * 16;

    v8f c = {};
    const float* arow = A + (size_t)(m0 + row) * HH;
    for (int k0 = 0; k0 < HH; k0 += 4) {
        const int ka = k0 + 2 * khi;
        v2f a, b;
        a.x = arow[ka];
        a.y = arow[ka + 1];
        b.x = Bm[(size_t)ka * HH + n0 + row];
        b.y = Bm[(size_t)(ka + 1) * HH + n0 + row];
        c = __builtin_amdgcn_wmma_f32_16x16x4_f32(false, a, false, b,
                                                  (short)0, c, false, false);
    }
#pragma unroll
    for (int v = 0; v < 8; ++v)
        D[(size_t)(m0 + v + 8 * khi) * HH + n0 + row] = c[v];
}

// ---------------------------------------------------------------------------
// Kernel 4: t_x[b] = sum_h xS[b,h] * prelogits[b,h]  (float4 loads)
// ---------------------------------------------------------------------------
__global__ void tx_kernel(const float4* __restrict__ xS4,     // [B,H/4]
                          const float4* __restrict__ pre4,    // [B,H/4]
                          float* __restrict__ t_x) {
    __shared__ float red[192];
    const int b = blockIdx.x;
    const int t = threadIdx.x;                 // 0..191
    const float4 x = xS4[(size_t)b * H4 + t];
    const float4 p = pre4[(size_t)b * H4 + t];
    red[t] = x.x * p.x + x.y * p.y + x.z * p.z + x.w * p.w;
    __syncthreads();

    if (t < 64) red[t] += red[t + 128];        // fold 192 -> 128
    __syncthreads();
    for (int off = 64; off >= 1; off >>= 1) {
        if (t < off) red[t] += red[t + off];
        __syncthreads();
    }
    if (t == 0) t_x[b] = red[0];
}

// ---------------------------------------------------------------------------
// Kernel 5: scores_over_tasks[c,b] = t_x[b] - 2*(xS @ means^T)[b,c] + t_m[c]
// fp32 WMMA with A = xS [B,H], B = means^T (B[k][n] = means[n][k]).
// C dim padded 150->160; B-operand row index clamped so EXEC stays all-1s
// through the WMMA loop (ISA requirement); invalid columns masked at store.
// Writes directly into d_out + B in [C,B] layout.
// ---------------------------------------------------------------------------
__global__ void scores_wmma_kernel(const float* __restrict__ xS,    // [B,H]
                                   const float* __restrict__ means, // [C,H]
                                   const float* __restrict__ t_x,   // [B]
                                   const float* __restrict__ t_m,   // [C]
                                   float* __restrict__ out_scores) {// [C,B]
    const int tile = blockIdx.x;               // 16 * 10 tiles
    const int tm = tile / (CPAD / 16);
    const int tn = tile % (CPAD / 16);
    const int lane = threadIdx.x;
    const int row  = lane & 15;
    const int khi  = lane >> 4;
    const int m0 = tm * 16, n0 = tn * 16;

    const int n  = n0 + row;                   // class index for this lane
    const int nc = (n < CC) ? n : (CC - 1);    // clamped: loads stay in-bounds,
                                               // EXEC stays full for WMMA
    v8f c = {};
    const float* arow = xS + (size_t)(m0 + row) * HH;
    const float* brow = means + (size_t)nc * HH;
    for (int k0 = 0; k0 < HH; k0 += 4) {
        const int ka = k0 + 2 * khi;
        v2f a, b;
        a.x = arow[ka];
        a.y = arow[ka + 1];
        b.x = brow[ka];
        b.y = brow[ka + 1];
        c = __builtin_amdgcn_wmma_f32_16x16x4_f32(false, a, false, b,
                                                  (short)0, c, false, false);
    }
    if (n < CC) {
        const float tmn = t_m[n];
#pragma unroll
        for (int v = 0; v < 8; ++v) {
            const int bidx = m0 + v + 8 * khi;
            out_scores[(size_t)n * BB + bidx] = t_x[bidx] - 2.0f * c[v] + tmn;
        }
    }
}

// ---------------------------------------------------------------------------
// Kernel 6: argmin over classes (first-min semantics like jnp.argmin)
// ---------------------------------------------------------------------------
__global__ void argmin_kernel(const float* __restrict__ scores, // [C,B]
                              float* __restrict__ out_idx) {    // [B]
    const int b = blockIdx.x * blockDim.x + threadIdx.x;
    if (b >= BB) return;
    float best = scores[b];
    int bi = 0;
    for (int c = 1; c < CC; ++c) {
        const float v = scores[(size_t)c * BB + b];
        if (v < best) { best = v; bi = c; }
    }
    out_idx[b] = (float)bi;
}

// ---------------------------------------------------------------------------
extern "C" void kernel_launch(void* const* d_in, const int* in_sizes, int n_in,
                              void* d_out, int out_size, void* d_ws, size_t ws_size,
                              hipStream_t stream) {
    const float* lhs    = (const float*)d_in[0];  // [B,S,H]
    const int*   mask   = (const int*)d_in[1];    // [B,S]
    const float* means  = (const float*)d_in[2];  // [C,H]
    const float* covinv = (const float*)d_in[3];  // [H,H]

    float* out = (float*)d_out;            // [0..B): indices, [B..B+C*B): scores [C,B]
    float* out_scores = out + BB;

    // workspace layout (floats)
    float* ws        = (float*)d_ws;
    float* prelogits = ws;                          // B*H
    float* xS        = prelogits + (size_t)BB * HH; // B*H
    float* t_x       = xS + (size_t)BB * HH;        // B
    float* t_m       = t_x + BB;                    // C

    pool_kernel<<<dim3(BB), dim3(H4), 0, stream>>>(
        (const float4*)lhs, mask, (float4*)prelogits);
    tm_kernel<<<dim3(CC), dim3(HH), 0, stream>>>(means, covinv, t_m);
    xs_wmma_kernel<<<dim3((BB / 16) * (HH / 16)), dim3(32), 0, stream>>>(
        prelogits, covinv, xS);
    tx_kernel<<<dim3(BB), dim3(H4), 0, stream>>>(
        (const float4*)xS, (const float4*)prelogits, t_x);
    scores_wmma_kernel<<<dim3((BB / 16) * (CPAD / 16)), dim3(32), 0, stream>>>(
        xS, means, t_x, t_m, out_scores);
    argmin_kernel<<<dim3(1), dim3(BB), 0, stream>>>(out_scores, out);
}